// ProcessParsingResult_15762529976933
// MI455X (gfx1250) — compile-verified
//
#include <hip/hip_runtime.h>
#include <math.h>

#define NCELL   3872      // 40^2+36^2+24^2+16^2+12^2
#define EDIM    128
#define NPIX    16384     // 128*128
#define NMSPRE  500
#define MAXIMG  30
#define BSTR    136       // B-tile LDS row stride in dwords (128 + 8 TDM pad)

typedef __attribute__((ext_vector_type(2))) float v2f;
typedef __attribute__((ext_vector_type(8))) float v8f;
typedef __attribute__((ext_vector_type(4))) unsigned int u32x4;
typedef __attribute__((ext_vector_type(8))) int i32x8;
typedef __attribute__((ext_vector_type(4))) int i32x4;

__device__ __forceinline__ float sigmoidf_(float x) {
  return 1.0f / (1.0f + __expf(-x));
}

__device__ __forceinline__ float cell_stride(int m) {
  if (m < 1600) return 4.f;
  if (m < 2896) return 8.f;
  if (m < 3472) return 16.f;
  if (m < 3728) return 32.f;
  return 64.f;
}

// Issue one TDM load: 128(K-rows) x 128(N-cols) f32 tile of B into LDS at lds_off.
// D# per CDNA5 ISA. Padding: +8 dwords every 128 dwords
// -> LDS row stride 136 dwords (bank-conflict-free WMMA fragment reads).
__device__ __forceinline__ void tdm_load_b_tile(unsigned lds_off, const float* gptr) {
  unsigned long long ga = (unsigned long long)(uintptr_t)gptr;
  u32x4 g0;
  g0.x = 1u;                                   // count=1 (valid), user mode, no gather
  g0.y = lds_off;                              // LDS byte address
  g0.z = (unsigned)(ga & 0xFFFFFFFFu);         // global_addr[31:0]
  g0.w = (unsigned)((ga >> 32) & 0x01FFFFFFu)  // global_addr[56:32]
         | (2u << 30);                         // type=2 ("image")
  i32x8 g1;
  g1[0] = (2 << 16)      // data_size = 4 bytes
        | (1 << 20)      // pad_enable
        | (6 << 22)      // pad_interval code 6 = every 128 dwords
        | (7 << 25);     // pad_amount  code 7 = 8 dwords
  g1[1] = (int)(16384u << 16);   // tensor_dim0[15:0]=16384 (atomic_barrier_addr=0)
  g1[2] = (128 << 16);           // tensor_dim0[31:16]=0 | tensor_dim1[15:0]=128
  g1[3] = (128 << 16);           // tensor_dim1[31:16]=0 | tile_dim0=128
  g1[4] = 128;                   // tile_dim1=128 | tile_dim2=0
  g1[5] = 16384;                 // tensor_dim0_stride[31:0]
  g1[6] = 0;                     // stride hi | tensor_dim1_stride lo (unused, 2D)
  g1[7] = 0;
  i32x4 gz = {0, 0, 0, 0};
#if __clang_major__ >= 23
  i32x8 gz8 = {0, 0, 0, 0, 0, 0, 0, 0};
  __builtin_amdgcn_tensor_load_to_lds(g0, g1, gz, gz, gz8, 0);
#else
  __builtin_amdgcn_tensor_load_to_lds(g0, g1, gz, gz, 0);
#endif
}

// ---------------------------------------------------------------- pack A + scores
__global__ __launch_bounds__(256) void pack_kernel(
    const float* c0, const float* k0, const float* c1, const float* k1,
    const float* c2, const float* k2, const float* c3, const float* k3,
    const float* c4, const float* k4, float* __restrict__ A, float* __restrict__ scores) {
  int idx = blockIdx.x * 256 + threadIdx.x;
  if (idx >= NCELL * EDIM) return;
  int m = idx >> 7;
  int e = idx & 127;
  const float* kp; const float* cp; int cell; int gg;
  if (m < 1600)      { kp = k0; cp = c0; cell = m;        gg = 1600; }
  else if (m < 2896) { kp = k1; cp = c1; cell = m - 1600; gg = 1296; }
  else if (m < 3472) { kp = k2; cp = c2; cell = m - 2896; gg = 576;  }
  else if (m < 3728) { kp = k3; cp = c3; cell = m - 3472; gg = 256;  }
  else               { kp = k4; cp = c4; cell = m - 3728; gg = 144;  }
  A[m * EDIM + e] = kp[e * gg + cell];
  if (e == 0) scores[m] = cp[cell];
}

// ------------------------------------------ fused WMMA GEMM + sigmoid + reductions + bitpack
// grid (16, 121), block 256 (8 waves). Block owns 32 M-rows x 1024 N-cols.
// B tiles (128K x 128N, 64KB) staged by the Tensor Data Mover into double-buffered
// LDS; wave w computes columns [w*16, w*16+16) of each staged chunk; K-loop of
// 32 x v_wmma_f32_16x16x4_f32 per 16x16 tile, 2 M-subtiles per wave per chunk.
__global__ __launch_bounds__(256) void gemm_masks_kernel(
    const float* __restrict__ A, const float* __restrict__ B,
    float* __restrict__ sum_cnt, float* __restrict__ sum_val,
    unsigned short* __restrict__ maskbits) {
  __shared__ float As[32][132];              // stride 132 -> conflict-free column reads
  __shared__ float Bs[2][128 * BSTR];        // two 69.6KB TDM-filled buffers
  const int mbase = blockIdx.y * 32;
  const int tid  = threadIdx.x;
  const int wave = tid >> 5;
  const int lane = tid & 31;
  const int row  = lane & 15;
  const int hlf  = lane >> 4;
  const unsigned lds0 = (unsigned)(uintptr_t)(void*)&Bs[0][0];
  const unsigned lds1 = (unsigned)(uintptr_t)(void*)&Bs[1][0];
  const float* gbase = B + (size_t)blockIdx.x * 1024;

  for (int i = tid; i < 32 * 128; i += 256) { // cooperative A-tile stage (16 KB)
    int r = i >> 7, c = i & 127;
    As[r][c] = A[(mbase + r) * EDIM + c];
  }
  if (wave == 0) tdm_load_b_tile(lds0, gbase);      // chunk 0 in flight

  float cnt[2][8]  = {};
  float msum[2][8] = {};
  __syncthreads();

  for (int it = 0; it < 8; ++it) {
    if (wave == 0) {
      if (it < 7) {                                  // prefetch chunk it+1
        tdm_load_b_tile((it & 1) ? lds0 : lds1, gbase + (it + 1) * 128);
        __builtin_amdgcn_s_wait_tensorcnt((short)1); // chunk it landed
      } else {
        __builtin_amdgcn_s_wait_tensorcnt((short)0);
      }
    }
    __syncthreads();                                 // publish chunk it to all waves
    const float* Bc = &Bs[it & 1][0];
    const int nloc = wave * 16;
    for (int mt = 0; mt < 2; ++mt) {
      v8f acc = {0.f, 0.f, 0.f, 0.f, 0.f, 0.f, 0.f, 0.f};
      for (int k0 = 0; k0 < EDIM; k0 += 4) {
        v2f a, b;
        a.x = As[mt * 16 + row][k0 + 2 * hlf];
        a.y = As[mt * 16 + row][k0 + 2 * hlf + 1];
        b.x = Bc[(k0 + 2 * hlf) * BSTR + nloc + row];
        b.y = Bc[(k0 + 2 * hlf + 1) * BSTR + nloc + row];
        acc = __builtin_amdgcn_wmma_f32_16x16x4_f32(
            false, a, false, b, (short)0, acc, false, false);
      }
      const int n0 = blockIdx.x * 1024 + it * 128 + nloc;
#pragma unroll
      for (int v = 0; v < 8; ++v) {
        float p = sigmoidf_(acc[v]);
        bool msk = p > 0.5f;
        cnt[mt][v]  += msk ? 1.0f : 0.0f;
        msum[mt][v] += msk ? p : 0.0f;
        unsigned int bits = (unsigned int)__ballot(msk ? 1 : 0);
        if (lane == 0)
          maskbits[(size_t)(mbase + mt * 16 + v) * 1024 + (n0 >> 4)] =
              (unsigned short)(bits & 0xFFFFu);
        if (lane == 16)
          maskbits[(size_t)(mbase + mt * 16 + v + 8) * 1024 + (n0 >> 4)] =
              (unsigned short)(bits >> 16);
      }
    }
    __syncthreads();            // buffer free before TDM refills it next iteration
  }
  // reduce over 16 lanes within each half-wave, then one atomic per row-half
#pragma unroll
  for (int mt = 0; mt < 2; ++mt) {
#pragma unroll
    for (int v = 0; v < 8; ++v) {
      float c = cnt[mt][v], s = msum[mt][v];
      for (int off = 1; off < 16; off <<= 1) {
        c += __shfl_xor(c, off, 32);
        s += __shfl_xor(s, off, 32);
      }
      if ((lane & 15) == 0) {
        int r = mbase + mt * 16 + v + 8 * hlf;
        atomicAdd(&sum_cnt[r], c);
        atomicAdd(&sum_val[r], s);
      }
    }
  }
}

// ---------------------------------------------------------------- cscores
__global__ __launch_bounds__(256) void cscore_kernel(
    const float* __restrict__ scores, const float* __restrict__ sum_cnt,
    const float* __restrict__ sum_val, float* __restrict__ cscores) {
  int m = blockIdx.x * 256 + threadIdx.x;
  if (m >= NCELL) return;
  float s = scores[m];
  float c = sum_cnt[m];
  bool keep = (s > 0.3f) && (c > cell_stride(m));
  float segscore = sum_val[m] / fmaxf(c, 1.0f);
  cscores[m] = keep ? s * segscore : 0.0f;
}

// ---------------------------------------------------------------- top-500 (bitonic 4096)
__global__ __launch_bounds__(1024) void sort4096_kernel(
    const float* __restrict__ cscores, const float* __restrict__ sum_cnt,
    float* __restrict__ top_scores, int* __restrict__ top_idx, float* __restrict__ sm_top) {
  __shared__ float sk[4096];
  __shared__ int   si[4096];
  int tid = threadIdx.x;
  for (int i = tid; i < 4096; i += 1024) {
    sk[i] = (i < NCELL) ? cscores[i] : -1e30f;
    si[i] = i;
  }
  __syncthreads();
  for (int k = 2; k <= 4096; k <<= 1)
    for (int j = k >> 1; j > 0; j >>= 1) {
      for (int i = tid; i < 4096; i += 1024) {
        int ixj = i ^ j;
        if (ixj > i) {
          bool desc = ((i & k) == 0);
          float a = sk[i], b = sk[ixj];
          if (desc ? (a < b) : (a > b)) {
            sk[i] = b; sk[ixj] = a;
            int t = si[i]; si[i] = si[ixj]; si[ixj] = t;
          }
        }
      }
      __syncthreads();
    }
  for (int i = tid; i < NMSPRE; i += 1024) {
    int m = si[i];
    top_scores[i] = sk[i];
    top_idx[i]    = m;
    sm_top[i]     = sum_cnt[m];
  }
}

// ---------------------------------------------------------------- IoU via bitpacked popcount
__global__ __launch_bounds__(512) void iou_kernel(
    const unsigned int* __restrict__ maskbits, const int* __restrict__ top_idx,
    const float* __restrict__ sm_top, float* __restrict__ iou) {
  __shared__ unsigned int rowi[512];
  int i = blockIdx.x;          // 0..499
  int j = threadIdx.x;
  const unsigned int* mi = maskbits + (size_t)top_idx[i] * 512;
  for (int w = j; w < 512; w += 512) rowi[w] = mi[w];
  __syncthreads();
  if (j < NMSPRE) {
    float v = 0.0f;
    if (j > i) {                // triu(k=1)
      const unsigned int* mj = maskbits + (size_t)top_idx[j] * 512;
      int inter = 0;
      for (int w = 0; w < 512; ++w) inter += __popc(rowi[w] & mj[w]);
      float un = sm_top[i] + sm_top[j] - (float)inter;
      v = (float)inter / fmaxf(un, 1.0f);
    }
    iou[i * 512 + j] = v;
  }
}

// ---------------------------------------------------------------- Matrix-NMS decay
__global__ __launch_bounds__(512) void compdecay_kernel(
    const float* __restrict__ iou, const float* __restrict__ top_scores,
    float* __restrict__ nms) {
  __shared__ float comp_s[512];
  int j = threadIdx.x;
  float mx = 0.0f;
  if (j < NMSPRE)
    for (int i = 0; i < NMSPRE; ++i) mx = fmaxf(mx, iou[i * 512 + j]);
  comp_s[j] = mx;
  __syncthreads();
  if (j < NMSPRE) {
    float best = -1e30f;
    for (int i = 0; i < NMSPRE; ++i) {
      float x = iou[i * 512 + j];
      float c = comp_s[i];
      best = fmaxf(best, x * x - c * c);
    }
    float coeff = __expf(-2.0f * best);   // min_i exp(-sigma*x_i) == exp(-sigma*max_i x_i)
    float ns = top_scores[j] * coeff;
    nms[j] = (ns >= 0.05f) ? ns : 0.0f;
  }
}

// ---------------------------------------------------------------- top-30 (bitonic 512)
__global__ __launch_bounds__(256) void sort512_kernel(
    const float* __restrict__ nms, const int* __restrict__ top_idx,
    float* __restrict__ d_out, int* __restrict__ sel, float* __restrict__ fsc) {
  __shared__ float sk[512];
  __shared__ int   si[512];
  int tid = threadIdx.x;
  for (int i = tid; i < 512; i += 256) {
    sk[i] = (i < NMSPRE) ? nms[i] : -1e30f;
    si[i] = i;
  }
  __syncthreads();
  for (int k = 2; k <= 512; k <<= 1)
    for (int j = k >> 1; j > 0; j >>= 1) {
      for (int i = tid; i < 512; i += 256) {
        int ixj = i ^ j;
        if (ixj > i) {
          bool desc = ((i & k) == 0);
          float a = sk[i], b = sk[ixj];
          if (desc ? (a < b) : (a > b)) {
            sk[i] = b; sk[ixj] = a;
            int t = si[i]; si[i] = si[ixj]; si[ixj] = t;
          }
        }
      }
      __syncthreads();
    }
  if (tid < MAXIMG) {
    float f = sk[tid];
    sel[tid] = top_idx[si[tid]];
    fsc[tid] = f;
    d_out[120 + tid] = f;                         // final_scores
    d_out[150 + tid] = (f > 0.3f) ? 1.0f : 0.0f;  // vis
  }
}

// ---------------------------------------------------------------- recompute 30 selected mask rows
__global__ __launch_bounds__(256) void selpred_kernel(
    const float* __restrict__ A, const float* __restrict__ B,
    const int* __restrict__ sel, float* __restrict__ sel_preds) {
  __shared__ float kr[EDIM];
  int i = blockIdx.x;
  int pix = blockIdx.y * 256 + threadIdx.x;
  int m = sel[i];
  if (threadIdx.x < EDIM) kr[threadIdx.x] = A[m * EDIM + threadIdx.x];
  __syncthreads();
  float acc = 0.0f;
  for (int e = 0; e < EDIM; ++e) acc += kr[e] * B[(size_t)e * NPIX + pix];
  sel_preds[(size_t)i * NPIX + pix] = sigmoidf_(acc);
}

// ---------------------------------------------------------------- bilinear 4x upsample + bbox
__global__ __launch_bounds__(256) void box_kernel(
    const float* __restrict__ sel_preds, const float* __restrict__ fsc,
    float* __restrict__ d_out) {
  __shared__ int rx0[256], rx1[256], ry0[256], ry1[256];
  int i = blockIdx.x;
  const float* P = sel_preds + (size_t)i * NPIX;
  int xmin = 512, xmax = -1, ymin = 512, ymax = -1;
  for (int p = threadIdx.x; p < 512 * 512; p += 256) {
    int y = p >> 9, x = p & 511;
    float sy = (y + 0.5f) * 0.25f - 0.5f;          // half-pixel centers, 128->512
    float sx = (x + 0.5f) * 0.25f - 0.5f;
    float fy0 = floorf(sy), fx0 = floorf(sx);
    float fy = sy - fy0, fx = sx - fx0;
    int y0 = (int)fy0, x0 = (int)fx0;
    int y1 = min(y0 + 1, 127), x1 = min(x0 + 1, 127);
    y0 = max(y0, 0); x0 = max(x0, 0);
    float v00 = P[y0 * 128 + x0], v01 = P[y0 * 128 + x1];
    float v10 = P[y1 * 128 + x0], v11 = P[y1 * 128 + x1];
    float val = (1.f - fy) * ((1.f - fx) * v00 + fx * v01) +
                fy * ((1.f - fx) * v10 + fx * v11);
    if (val > 0.5f) {
      xmin = min(xmin, x); xmax = max(xmax, x);
      ymin = min(ymin, y); ymax = max(ymax, y);
    }
  }
  rx0[threadIdx.x] = xmin; rx1[threadIdx.x] = xmax;
  ry0[threadIdx.x] = ymin; ry1[threadIdx.x] = ymax;
  __syncthreads();
  for (int s = 128; s > 0; s >>= 1) {
    if (threadIdx.x < s) {
      rx0[threadIdx.x] = min(rx0[threadIdx.x], rx0[threadIdx.x + s]);
      rx1[threadIdx.x] = max(rx1[threadIdx.x], rx1[threadIdx.x + s]);
      ry0[threadIdx.x] = min(ry0[threadIdx.x], ry0[threadIdx.x + s]);
      ry1[threadIdx.x] = max(ry1[threadIdx.x], ry1[threadIdx.x + s]);
    }
    __syncthreads();
  }
  if (threadIdx.x == 0) {
    bool vis = fsc[i] > 0.3f;
    d_out[i * 4 + 0] = vis ? (float)rx0[0] : 0.f;
    d_out[i * 4 + 1] = vis ? (float)ry0[0] : 0.f;
    d_out[i * 4 + 2] = vis ? (float)rx1[0] : 0.f;
    d_out[i * 4 + 3] = vis ? (float)ry1[0] : 0.f;
  }
}

// ----------------------------------------------------------------
extern "C" void kernel_launch(void* const* d_in, const int* in_sizes, int n_in,
                              void* d_out, int out_size, void* d_ws, size_t ws_size,
                              hipStream_t stream) {
  (void)in_sizes; (void)n_in; (void)out_size; (void)ws_size;
  // setup_inputs() dict order: cate0,kern0,cate1,kern1,...,cate4,kern4,seg_pred
  const float* c0 = (const float*)d_in[0];
  const float* k0 = (const float*)d_in[1];
  const float* c1 = (const float*)d_in[2];
  const float* k1 = (const float*)d_in[3];
  const float* c2 = (const float*)d_in[4];
  const float* k2 = (const float*)d_in[5];
  const float* c3 = (const float*)d_in[6];
  const float* k3 = (const float*)d_in[7];
  const float* c4 = (const float*)d_in[8];
  const float* k4 = (const float*)d_in[9];
  const float* seg = (const float*)d_in[10];

  char* ws = (char*)d_ws;
  size_t off = 0;
  auto alloc = [&](size_t b) -> void* {
    void* p = ws + off;
    off = (off + b + 255) & ~(size_t)255;
    return p;
  };
  float* A          = (float*)alloc((size_t)NCELL * EDIM * 4);
  float* scores     = (float*)alloc((size_t)NCELL * 4);
  float* sum_cnt    = (float*)alloc((size_t)NCELL * 4);
  float* sum_val    = (float*)alloc((size_t)NCELL * 4);
  float* cscores    = (float*)alloc((size_t)NCELL * 4);
  unsigned short* maskbits = (unsigned short*)alloc((size_t)NCELL * 2048);
  float* top_scores = (float*)alloc(512 * 4);
  int*   top_idx    = (int*)alloc(512 * 4);
  float* sm_top     = (float*)alloc(512 * 4);
  float* iou        = (float*)alloc((size_t)NMSPRE * 512 * 4);
  float* nms        = (float*)alloc(512 * 4);
  int*   sel        = (int*)alloc(32 * 4);
  float* fsc        = (float*)alloc(32 * 4);
  float* sel_preds  = (float*)alloc((size_t)MAXIMG * NPIX * 4);

  hipMemsetAsync(sum_cnt, 0, (size_t)NCELL * 4, stream);
  hipMemsetAsync(sum_val, 0, (size_t)NCELL * 4, stream);

  pack_kernel<<<(NCELL * EDIM) / 256, 256, 0, stream>>>(
      c0, k0, c1, k1, c2, k2, c3, k3, c4, k4, A, scores);
  gemm_masks_kernel<<<dim3(16, 121), 256, 0, stream>>>(
      A, seg, sum_cnt, sum_val, maskbits);
  cscore_kernel<<<(NCELL + 255) / 256, 256, 0, stream>>>(
      scores, sum_cnt, sum_val, cscores);
  sort4096_kernel<<<1, 1024, 0, stream>>>(cscores, sum_cnt, top_scores, top_idx, sm_top);
  iou_kernel<<<NMSPRE, 512, 0, stream>>>((const unsigned int*)maskbits, top_idx, sm_top, iou);
  compdecay_kernel<<<1, 512, 0, stream>>>(iou, top_scores, nms);
  sort512_kernel<<<1, 256, 0, stream>>>(nms, top_idx, (float*)d_out, sel, fsc);
  selpred_kernel<<<dim3(MAXIMG, NPIX / 256), 256, 0, stream>>>(A, seg, sel, sel_preds);
  box_kernel<<<MAXIMG, 256, 0, stream>>>(sel_preds, fsc, (float*)d_out);
}